// LocalGraph_47270410060163
// MI455X (gfx1250) — compile-verified
//
#include <hip/hip_runtime.h>
#include <math.h>

#define N_NODES 100000
#define N_EDGES 3200000
#define DIM 64
#define NUM_CANDS 1024
#define CHUNK 64            // edges staged per async copy (512B: 256B cols + 256B vals)

typedef float v2f __attribute__((ext_vector_type(2)));
typedef float v8f __attribute__((ext_vector_type(8)));

// ---------------- threefry-2x32 (matches jax.random) ----------------
__host__ __device__ inline unsigned rotl32(unsigned v, int n) {
  return (v << n) | (v >> (32 - n));
}

__host__ __device__ inline void tf2x32(unsigned k0, unsigned k1,
                                       unsigned x0, unsigned x1,
                                       unsigned& o0, unsigned& o1) {
  const unsigned ks2 = k0 ^ k1 ^ 0x1BD11BDAu;
  const int ra[4] = {13, 15, 26, 6}, rb[4] = {17, 29, 16, 24};
  x0 += k0; x1 += k1;
  for (int i = 0; i < 4; ++i) { x0 += x1; x1 = rotl32(x1, ra[i]); x1 ^= x0; }
  x0 += k1; x1 += ks2 + 1u;
  for (int i = 0; i < 4; ++i) { x0 += x1; x1 = rotl32(x1, rb[i]); x1 ^= x0; }
  x0 += ks2; x1 += k0 + 2u;
  for (int i = 0; i < 4; ++i) { x0 += x1; x1 = rotl32(x1, ra[i]); x1 ^= x0; }
  x0 += k0; x1 += k1 + 3u;
  for (int i = 0; i < 4; ++i) { x0 += x1; x1 = rotl32(x1, rb[i]); x1 ^= x0; }
  x0 += k1; x1 += ks2 + 4u;
  for (int i = 0; i < 4; ++i) { x0 += x1; x1 = rotl32(x1, ra[i]); x1 ^= x0; }
  x0 += ks2; x1 += k0 + 5u;
  o0 = x0; o1 = x1;
}

// jax-style uniform [0,1): element i of an n-element draw (n even)
__device__ inline float tf_uniform(unsigned k0, unsigned k1, unsigned i, unsigned n) {
  const unsigned half = n >> 1;
  unsigned o0, o1, bits;
  if (i < half) { tf2x32(k0, k1, i, i + half, o0, o1); bits = o0; }
  else          { tf2x32(k0, k1, i - half, i, o0, o1); bits = o1; }
  return __builtin_bit_cast(float, (bits >> 9) | 0x3f800000u) - 1.0f;
}

// ---------------- CDNA5 async global->LDS helpers ----------------
__device__ inline void async_copy_b128(unsigned lds_dst, unsigned long long gsrc) {
  // GLOBAL_LOAD_ASYNC_TO_LDS_B128: per-lane 16B memory -> LDS, tracked by ASYNCcnt
  asm volatile("global_load_async_to_lds_b128 %0, %1, off"
               :: "v"(lds_dst), "v"(gsrc) : "memory");
}
__device__ inline void wait_async0() {
  asm volatile("s_wait_asynccnt 0x0" ::: "memory");
}
__device__ inline void wait_async1() {
  asm volatile("s_wait_asynccnt 0x1" ::: "memory");
}

// ---------------- CSR build ----------------
__global__ void k_degree(const int* __restrict__ rows, int* __restrict__ deg, int e) {
  const int i = blockIdx.x * blockDim.x + threadIdx.x;
  if (i < e) atomicAdd(&deg[rows[i]], 1);
}

__global__ void __launch_bounds__(1024)
k_scan(const int* __restrict__ deg, int* __restrict__ row_off,
       int* __restrict__ cursor, int n, int total) {
  __shared__ int buf[1024];
  __shared__ int carry;
  const int tid = threadIdx.x;
  if (tid == 0) carry = 0;
  __syncthreads();
  for (int base = 0; base < n; base += 1024) {
    const int i = base + tid;
    const int v = (i < n) ? deg[i] : 0;
    buf[tid] = v;
    __syncthreads();
    for (int o = 1; o < 1024; o <<= 1) {
      const int t = (tid >= o) ? buf[tid - o] : 0;
      __syncthreads();
      buf[tid] += t;
      __syncthreads();
    }
    const int c = carry;
    const int excl = buf[tid] - v;
    if (i < n) { row_off[i] = c + excl; cursor[i] = c + excl; }
    __syncthreads();
    if (tid == 1023) carry = c + buf[1023];
    __syncthreads();
  }
  if (tid == 0) row_off[n] = total;
}

// scatter + precompute dropout-masked edge values per depth (raw bytes so the
// SPMM passes can stage them with async copies, no per-edge mask math there)
__global__ void k_scatter(const int* __restrict__ rows, const int* __restrict__ cols,
                          const float* __restrict__ vals, int* __restrict__ cursor,
                          int* __restrict__ csr_col, float* __restrict__ v0,
                          float* __restrict__ v1, float* __restrict__ v2,
                          unsigned kd1_0, unsigned kd1_1,
                          unsigned kd2_0, unsigned kd2_1, int e) {
  const int i = blockIdx.x * blockDim.x + threadIdx.x;
  if (i >= e) return;
  const int r = rows[i];
  const int pos = atomicAdd(&cursor[r], 1);
  csr_col[pos] = cols[i];
  const float v = vals[i];
  // dropout masks from the ORIGINAL edge index (matches reference semantics)
  const float u1 = tf_uniform(kd1_0, kd1_1, (unsigned)i, (unsigned)e);
  const float u2 = tf_uniform(kd2_0, kd2_1, (unsigned)i, (unsigned)e);
  const float m1 = (u1 >= 0.5f)  ? 1.f : 0.f;   // keep prob 0.5  (floor(u+0.5))
  const float m2 = (u2 >= 0.75f) ? 1.f : 0.f;   // keep prob 0.25 (floor(u+0.25))
  v0[pos] = v;
  v1[pos] = v * m1;
  v2[pos] = v * m1 * m2;
}

// ------- fused SPMM pass: async LDS edge staging, one wave32 per row --------
__global__ void __launch_bounds__(256)
k_pass(const int* __restrict__ row_off, const int* __restrict__ csr_col,
       const float* __restrict__ csr_val, const float* __restrict__ xin,
       const float* __restrict__ numin, const float* __restrict__ orderin,
       float* __restrict__ xnew, float* __restrict__ numout,
       float* __restrict__ orderout, float* __restrict__ emb_sum,
       float* __restrict__ num_sum, int first, int n) {
  __shared__ char smem[8 * 1024];                 // 8 waves x 2 x 512B buffers
  const int lane = threadIdx.x & 31;
  const int wrp  = threadIdx.x >> 5;
  const int r = (blockIdx.x * blockDim.x + threadIdx.x) >> 5;
  if (r >= n) return;
  char* my = smem + wrp * 1024;
  // low 32 bits of the generic pointer are the LDS byte address (aperture layout)
  const unsigned lds0 = (unsigned)(unsigned long long)(void*)my;
  // per-lane async staging addresses: lanes 0-15 copy cols, lanes 16-31 copy vals
  const unsigned lane16 = (unsigned)(lane & 15) * 16u;
  const unsigned dst_off = (lane < 16) ? lane16 : (256u + lane16);

  const int s = row_off[r], e = row_off[r + 1];
  float accx = 0.f, accy = 0.f, vsum = 0.f, nsum = 0.f;

  if (s < e) {
    {  // stage chunk 0 into buffer 0
      const unsigned long long src = (lane < 16)
          ? ((unsigned long long)(const char*)(csr_col + s) + lane16)
          : ((unsigned long long)(const char*)(csr_val + s) + lane16);
      async_copy_b128(lds0 + dst_off, src);
    }
    int b = 0;
    for (int base = s; base < e; base += CHUNK) {
      const int next = base + CHUNK;
      if (next < e) {  // prefetch next chunk into the other buffer
        const unsigned long long src = (lane < 16)
            ? ((unsigned long long)(const char*)(csr_col + next) + lane16)
            : ((unsigned long long)(const char*)(csr_val + next) + lane16);
        async_copy_b128(lds0 + (unsigned)(1 - b) * 512u + dst_off, src);
        wait_async1();          // oldest (current chunk) complete, prefetch in flight
      } else {
        wait_async0();          // drain
      }
      const int* cbuf = (const int*)(my + b * 512);
      const float* vbuf = (const float*)(my + b * 512 + 256);
      const int cnt = min(CHUNK, e - base);
      for (int j = 0; j < cnt; ++j) {
        const float v = vbuf[j];        // broadcast LDS read (uniform address)
        vsum += v;
        if (v != 0.f) {
          const int cj = cbuf[j];
          if (!first) nsum += v * numin[cj];
          const float2 xr = ((const float2*)(xin + (size_t)cj * DIM))[lane];
          accx = fmaf(v, xr.x, accx);
          accy = fmaf(v, xr.y, accy);
        }
      }
      b ^= 1;
    }
  }
  // every lane holds identical vsum/nsum (uniform edge loop) -> no reduction
  const float2 xs = ((const float2*)(xin + (size_t)r * DIM))[lane];
  const float ord = first ? 0.f : orderin[r];
  float2 en;
  en.x = accx - xs.x - ord * xs.x;
  en.y = accy - xs.y - ord * xs.y;
  ((float2*)(xnew + (size_t)r * DIM))[lane] = en;
  if (first) {
    ((float2*)(emb_sum + (size_t)r * DIM))[lane] = en;
  } else {
    float2 s2 = ((const float2*)(emb_sum + (size_t)r * DIM))[lane];
    s2.x += en.x; s2.y += en.y;
    ((float2*)(emb_sum + (size_t)r * DIM))[lane] = s2;
  }
  if (lane == 0) {
    orderout[r] = vsum;
    if (first) { numout[r] = vsum; num_sum[r] = vsum; }
    else {
      const float nn = nsum - numin[r] - ord;
      numout[r] = nn;
      num_sum[r] += nn;
    }
  }
}

// ---------------- row L2-normalize (subgraph and embeds) ----------------
__global__ void __launch_bounds__(256)
k_norm(const float* __restrict__ emb_sum, const float* __restrict__ num_sum,
       const float* __restrict__ embeds, float* __restrict__ subn,
       float* __restrict__ embn, int n) {
  const int lane = threadIdx.x & 31;
  const int r = (blockIdx.x * blockDim.x + threadIdx.x) >> 5;
  if (r >= n) return;
  const size_t base = (size_t)r * DIM;
  const float ns = num_sum[r] + 1e-8f;
  const float2 e2 = ((const float2*)(emb_sum + base))[lane];
  float2 t; t.x = e2.x / ns; t.y = e2.y / ns;
  float sq = t.x * t.x + t.y * t.y;
  for (int m = 16; m >= 1; m >>= 1) sq += __shfl_xor(sq, m);
  const float inv = 1.0f / fmaxf(sqrtf(sq), 1e-12f);
  float2 o; o.x = t.x * inv; o.y = t.y * inv;
  ((float2*)(subn + base))[lane] = o;

  const float2 b2 = ((const float2*)(embeds + base))[lane];
  float sq2 = b2.x * b2.x + b2.y * b2.y;
  for (int m = 16; m >= 1; m >>= 1) sq2 += __shfl_xor(sq2, m);
  const float inv2 = 1.0f / fmaxf(sqrtf(sq2), 1e-12f);
  float2 o2; o2.x = b2.x * inv2; o2.y = b2.y * inv2;
  ((float2*)(embn + base))[lane] = o2;
}

// ---------------- scores via WMMA f32 16x16x4 (diag of A*B) + gumbel -------
__global__ void __launch_bounds__(256)
k_dot_wmma(const float* __restrict__ sub, const float* __restrict__ embn,
           float* __restrict__ scores, unsigned kn0, unsigned kn1, int n) {
  const int lane = threadIdx.x & 31;
  const int wid = (blockIdx.x * blockDim.x + threadIdx.x) >> 5;
  const int tiles = n / 16;
  if (wid >= tiles) return;            // wave-uniform; EXEC stays all-ones
  const int r0 = wid * 16;
  const int mrow = lane & 15;          // lanes 0-15: M=lane (K 0,1); 16-31: M=lane-16 (K 2,3)
  const int kofs = (lane < 16) ? 0 : 2;
  const float* arow = sub  + (size_t)(r0 + mrow) * DIM + kofs;
  const float* brow = embn + (size_t)(r0 + mrow) * DIM + kofs;  // B: N=lane, same mapping
  v8f c = {0.f, 0.f, 0.f, 0.f, 0.f, 0.f, 0.f, 0.f};
#pragma unroll
  for (int kk = 0; kk < DIM; kk += 4) {
    v2f a, b;
    a[0] = arow[kk]; a[1] = arow[kk + 1];
    b[0] = brow[kk]; b[1] = brow[kk + 1];
    c = __builtin_amdgcn_wmma_f32_16x16x4_f32(false, a, false, b,
                                              (short)0, c, false, false);
  }
  // diagonal (m,m): m<8 -> lane m, vgpr m; m>=8 -> lane m+16, vgpr m-8
  const int sel = (lane < 8) ? lane : (lane >= 24 ? lane - 24 : -1);
  float d = 0.f;
#pragma unroll
  for (int q = 0; q < 8; ++q) if (sel == q) d = c[q];
  const int row = (lane < 8) ? (r0 + lane) : (lane >= 24 ? (r0 + lane - 16) : -1);
  if (row >= 0) {
    const float u = tf_uniform(kn0, kn1, (unsigned)row, (unsigned)n);
    const float g = -__logf(-__logf(u));      // gumbel
    scores[row] = d + g;
  }
}

// ---------------- top-k: radix-select threshold + LDS bitonic sort ---------
__global__ void __launch_bounds__(1024)
k_topk(const float* __restrict__ scores, float* __restrict__ cand, int n, int K) {
  __shared__ unsigned hist[256];
  __shared__ unsigned long long arr[2048];
  __shared__ unsigned sh_prefix, sh_rem, sh_cnt;
  const int tid = threadIdx.x;
  if (tid == 0) { sh_prefix = 0u; sh_rem = (unsigned)K; sh_cnt = 0u; }
  __syncthreads();
  for (int p = 3; p >= 0; --p) {
    if (tid < 256) hist[tid] = 0u;
    __syncthreads();
    const unsigned prefix = sh_prefix;
    const unsigned pmask = (p == 3) ? 0u : (0xFFFFFFFFu << (unsigned)((p + 1) * 8));
    const int shft = p * 8;
    for (int i = tid; i < n; i += 1024) {
      unsigned u = __float_as_uint(scores[i]);
      u = (u & 0x80000000u) ? ~u : (u | 0x80000000u);   // order-preserving key
      if ((u & pmask) == prefix) atomicAdd(&hist[(u >> shft) & 255u], 1u);
    }
    __syncthreads();
    if (tid == 0) {
      unsigned rem = sh_rem;
      int b = 255;
      for (; b > 0; --b) {
        if (hist[b] < rem) rem -= hist[b];
        else break;
      }
      sh_prefix = prefix | ((unsigned)b << shft);
      sh_rem = rem;
    }
    __syncthreads();
  }
  const unsigned T = sh_prefix;   // key of K-th largest element
  for (int i = tid; i < 2048; i += 1024) arr[i] = 0ull;
  __syncthreads();
  for (int i = tid; i < n; i += 1024) {
    unsigned u = __float_as_uint(scores[i]);
    u = (u & 0x80000000u) ? ~u : (u | 0x80000000u);
    if (u >= T) {
      const unsigned pos = atomicAdd(&sh_cnt, 1u);
      if (pos < 2048u)
        arr[pos] = ((unsigned long long)u << 32) | (unsigned)(~(unsigned)i);
    }
  }
  __syncthreads();
  // bitonic sort, descending; ~idx in low word => equal keys sort by idx asc
  for (int k2 = 2; k2 <= 2048; k2 <<= 1) {
    for (int j = k2 >> 1; j > 0; j >>= 1) {
      for (int i = tid; i < 2048; i += 1024) {
        const int ixj = i ^ j;
        if (ixj > i) {
          const bool desc = ((i & k2) == 0);
          const unsigned long long x = arr[i], y = arr[ixj];
          if (desc ? (x < y) : (x > y)) { arr[i] = y; arr[ixj] = x; }
        }
      }
      __syncthreads();
    }
  }
  for (int i = tid; i < K; i += 1024)
    cand[i] = (float)(~(unsigned)(arr[i] & 0xFFFFFFFFull));
}

// ---------------- launcher ----------------
extern "C" void kernel_launch(void* const* d_in, const int* in_sizes, int n_in,
                              void* d_out, int out_size, void* d_ws, size_t ws_size,
                              hipStream_t stream) {
  const int*   edge_rows = (const int*)d_in[0];
  const int*   edge_cols = (const int*)d_in[1];
  const float* edge_vals = (const float*)d_in[2];
  const float* embeds    = (const float*)d_in[3];
  const int N = N_NODES, E = N_EDGES;
  (void)in_sizes; (void)n_in; (void)out_size; (void)ws_size;

  char* w = (char*)d_ws;
  size_t off = 0;
  auto alloc = [&](size_t bytes) -> void* {
    void* p = w + off;
    off = (off + bytes + 255) & ~(size_t)255;
    return p;
  };
  const size_t Epad = (size_t)E + CHUNK;   // padding so full-chunk staging is in-bounds
  int*   deg      = (int*)alloc((size_t)N * 4);
  int*   row_off  = (int*)alloc((size_t)(N + 1) * 4);
  int*   cursor   = (int*)alloc((size_t)N * 4);
  int*   csr_col  = (int*)alloc(Epad * 4);
  float* csr_v0   = (float*)alloc(Epad * 4);
  float* csr_v1   = (float*)alloc(Epad * 4);
  float* csr_v2   = (float*)alloc(Epad * 4);
  float* emb_a    = (float*)alloc((size_t)N * DIM * 4);
  float* emb_b    = (float*)alloc((size_t)N * DIM * 4);
  float* emb_sum  = (float*)alloc((size_t)N * DIM * 4);
  float* num_a    = (float*)alloc((size_t)N * 4);
  float* num_b    = (float*)alloc((size_t)N * 4);
  float* num_sum  = (float*)alloc((size_t)N * 4);
  float* order_a  = (float*)alloc((size_t)N * 4);
  float* order_b  = (float*)alloc((size_t)N * 4);

  // host-side threefry key schedule mirroring jax.random.split from key(42)
  unsigned a0, a1, b0, b1;
  unsigned k0 = 0u, k1 = 42u;
  tf2x32(k0, k1, 0u, 2u, a0, a1); tf2x32(k0, k1, 1u, 3u, b0, b1);
  const unsigned key1_0 = a0, key1_1 = b0, kd1_0 = a1, kd1_1 = b1;   // depth 1
  tf2x32(key1_0, key1_1, 0u, 2u, a0, a1); tf2x32(key1_0, key1_1, 1u, 3u, b0, b1);
  const unsigned key2_0 = a0, key2_1 = b0, kd2_0 = a1, kd2_1 = b1;   // depth 2
  tf2x32(key2_0, key2_1, 0u, 2u, a0, a1); tf2x32(key2_0, key2_1, 1u, 3u, b0, b1);
  const unsigned kn0 = a1, kn1 = b1;                                  // gumbel noise

  hipMemsetAsync(deg, 0, (size_t)N * 4, stream);
  k_degree<<<(E + 255) / 256, 256, 0, stream>>>(edge_rows, deg, E);
  k_scan<<<1, 1024, 0, stream>>>(deg, row_off, cursor, N, E);
  k_scatter<<<(E + 255) / 256, 256, 0, stream>>>(edge_rows, edge_cols, edge_vals,
                                                 cursor, csr_col, csr_v0, csr_v1,
                                                 csr_v2, kd1_0, kd1_1, kd2_0, kd2_1, E);
  const int rowBlocks = (N + 7) / 8;   // one wave32 per row, 8 waves / 256-block
  k_pass<<<rowBlocks, 256, 0, stream>>>(row_off, csr_col, csr_v0,
                                        embeds, nullptr, nullptr,
                                        emb_a, num_a, order_a, emb_sum, num_sum, 1, N);
  k_pass<<<rowBlocks, 256, 0, stream>>>(row_off, csr_col, csr_v1,
                                        emb_a, num_a, order_a,
                                        emb_b, num_b, order_b, emb_sum, num_sum, 0, N);
  k_pass<<<rowBlocks, 256, 0, stream>>>(row_off, csr_col, csr_v2,
                                        emb_b, num_b, order_b,
                                        emb_a, num_a, order_a, emb_sum, num_sum, 0, N);
  k_norm<<<rowBlocks, 256, 0, stream>>>(emb_sum, num_sum, embeds, emb_a, emb_b, N);
  const int tiles = N / 16;
  const int dotBlocks = (tiles * 32 + 255) / 256;
  k_dot_wmma<<<dotBlocks, 256, 0, stream>>>(emb_a, emb_b, (float*)d_out, kn0, kn1, N);
  k_topk<<<1, 1024, 0, stream>>>((const float*)d_out, (float*)d_out + N, N, NUM_CANDS);
}